// SSIMLoss_50714973831260
// MI455X (gfx1250) — compile-verified
//
#include <hip/hip_runtime.h>

// SSIM loss, fused single pass for (32,3,512,512) fp32 on gfx1250 (wave32).
// 11x11 separable box filter expressed as banded-Toeplitz matmuls executed
// with V_WMMA_F32_16X16X4_F32; tile staging via global_load_async_to_lds_b32.

#define IMG_H 512
#define IMG_W 512
#define NPLANES 96            // N*C = 32*3
#define TILE_Y 16
#define TILE_X 64
#define PADR 28               // 16+10 halo rows, padded to 28 (7 k-steps * 4)
#define LDW 80                // LDS row stride (64+10 cols, padded)
#define TOTAL_PIX 25165824.0  // 32*3*512*512

typedef __attribute__((ext_vector_type(2))) float v2f;
typedef __attribute__((ext_vector_type(8))) float v8f;

__device__ __forceinline__ v8f wmma16x16x4(v2f a, v2f b, v8f c) {
  // (neg_a, A, neg_b, B, c_mod, C, reuse_a, reuse_b)
  return __builtin_amdgcn_wmma_f32_16x16x4_f32(false, a, false, b, (short)0, c,
                                               false, false);
}

__global__ __launch_bounds__(256) void ssim_tile_kernel(
    const float* __restrict__ pred, const float* __restrict__ targ,
    double* __restrict__ acc) {
  __shared__ float sp[PADR * LDW];        // padded pred tile
  __shared__ float st[PADR * LDW];        // padded target tile
  __shared__ float sv[5 * TILE_Y * LDW];  // vertically filtered fields
  __shared__ float red[256];

  const int tid = threadIdx.x;
  const int lane = tid & 31;
  // Wave index as a compiler-provable scalar (SGPR) so wave scheduling uses
  // scalar branches instead of EXEC-mask divergence around the WMMAs.
  const int swave = __builtin_amdgcn_readfirstlane(tid) >> 5;

  const int plane = blockIdx.z;
  const int y0 = blockIdx.y * TILE_Y;
  const int x0 = blockIdx.x * TILE_X;
  const float* __restrict__ P = pred + (size_t)plane * (IMG_H * IMG_W);
  const float* __restrict__ T = targ + (size_t)plane * (IMG_H * IMG_W);

  // ---- Phase 0: async global->LDS tile staging with zero halo ----
  for (int i = tid; i < PADR * LDW; i += 256) {
    const int r = i / LDW;
    const int c = i - r * LDW;
    const int gy = y0 - 5 + r;
    const int gx = x0 - 5 + c;
    const bool ok = (r < 26) && (c < 74) && ((unsigned)gy < (unsigned)IMG_H) &&
                    ((unsigned)gx < (unsigned)IMG_W);
    if (ok) {
      const unsigned goff = (unsigned)(gy * IMG_W + gx) * 4u;
      // Flat-shared address low 32 bits == LDS byte offset (aperture rule).
      const unsigned lp = (unsigned)(uintptr_t)(const void*)&sp[i];
      const unsigned lt = (unsigned)(uintptr_t)(const void*)&st[i];
      asm volatile("global_load_async_to_lds_b32 %0, %1, %2"
                   :: "v"(lp), "v"(goff), "s"(P) : "memory");
      asm volatile("global_load_async_to_lds_b32 %0, %1, %2"
                   :: "v"(lt), "v"(goff), "s"(T) : "memory");
    } else {
      sp[i] = 0.0f;  // zero padding: emulates count_include_pad and the
      st[i] = 0.0f;  // k-overrun rows read by the 7th WMMA k-step
    }
  }
  asm volatile("s_wait_asynccnt 0x0" ::: "memory");
  __syncthreads();

  // Lane decomposition shared by both WMMA phases (ISA 16x16x4 f32 layouts).
  const int m = lane & 15;             // A row / B-N / D column index
  const int khalf = (lane >> 4) << 1;  // lanes 16-31 hold K+2, K+3

  // ---- Phase 1: vertical box filter via WMMA (banded filter x data) ----
  // Waves 0..4 each own one 16-column group; all 5 fields accumulated
  // together so p/t are loaded from LDS once per k-step (branch-free).
  if (swave < 5) {
    const int col = swave * 16 + m;
    v8f D[5];
#pragma unroll
    for (int q = 0; q < 5; ++q) D[q] = (v8f){0.f, 0.f, 0.f, 0.f, 0.f, 0.f, 0.f, 0.f};
#pragma unroll
    for (int s = 0; s < 7; ++s) {
      const int kb = 4 * s + khalf;
      v2f a;
      a.x = ((unsigned)(kb - m) <= 10u) ? (1.0f / 11.0f) : 0.0f;
      a.y = ((unsigned)(kb + 1 - m) <= 10u) ? (1.0f / 11.0f) : 0.0f;
      const float p0 = sp[kb * LDW + col];
      const float p1 = sp[(kb + 1) * LDW + col];
      const float t0 = st[kb * LDW + col];
      const float t1 = st[(kb + 1) * LDW + col];
      const v2f b0 = {p0, p1};
      const v2f b1 = {t0, t1};
      const v2f b2 = {p0 * p0, p1 * p1};
      const v2f b3 = {t0 * t0, t1 * t1};
      const v2f b4 = {p0 * t0, p1 * t1};
      D[0] = wmma16x16x4(a, b0, D[0]);
      D[1] = wmma16x16x4(a, b1, D[1]);
      D[2] = wmma16x16x4(a, b2, D[2]);
      D[3] = wmma16x16x4(a, b3, D[3]);
      D[4] = wmma16x16x4(a, b4, D[4]);
    }
    const int rbase = (lane >> 4) << 3;  // D rows: j (lanes<16) / j+8
#pragma unroll
    for (int q = 0; q < 5; ++q) {
      float* __restrict__ dst = &sv[q * (TILE_Y * LDW)];
#pragma unroll
      for (int j = 0; j < 8; ++j) dst[(rbase + j) * LDW + col] = D[q][j];
    }
  }
  __syncthreads();

  // ---- Phase 2: horizontal box filter via WMMA (data x banded filter),
  //      SSIM map fused in registers ----
  float local = 0.0f;
  if (swave < 4) {
    const int xg = swave * 16;  // this wave's 16-column output group
    v8f D[5];
#pragma unroll
    for (int q = 0; q < 5; ++q) {
      const float* __restrict__ vq = &sv[q * (TILE_Y * LDW)];
      v8f d = {0.f, 0.f, 0.f, 0.f, 0.f, 0.f, 0.f, 0.f};
#pragma unroll
      for (int s = 0; s < 7; ++s) {
        const int kb = xg + 4 * s + khalf;
        v2f a, b;
        a.x = vq[m * LDW + kb];
        a.y = vq[m * LDW + kb + 1];
        const int rel = kb - xg - m;
        b.x = ((unsigned)rel <= 10u) ? (1.0f / 11.0f) : 0.0f;
        b.y = ((unsigned)(rel + 1) <= 10u) ? (1.0f / 11.0f) : 0.0f;
        d = wmma16x16x4(a, b, d);
      }
      D[q] = d;
    }
    const float C1 = 0.0001f;  // 0.01^2
    const float C2 = 0.0009f;  // 0.03^2
#pragma unroll
    for (int j = 0; j < 8; ++j) {
      const float mu1 = D[0][j], mu2 = D[1][j];
      const float e11 = D[2][j], e22 = D[3][j], e12 = D[4][j];
      const float m11 = mu1 * mu1, m22 = mu2 * mu2, m12 = mu1 * mu2;
      const float s11 = e11 - m11, s22 = e22 - m22, s12 = e12 - m12;
      const float num = (2.0f * m12 + C1) * (2.0f * s12 + C2);
      const float den = (m11 + m22 + C1) * (s11 + s22 + C2);
      local += num / den;
    }
  }

  // ---- Phase 3: block reduction + one f64 atomic ----
  red[tid] = local;
  __syncthreads();
#pragma unroll
  for (int off = 128; off > 0; off >>= 1) {
    if (tid < off) red[tid] += red[tid + off];
    __syncthreads();
  }
  if (tid == 0) atomicAdd(acc, (double)red[0]);
}

__global__ void ssim_finalize(const double* __restrict__ acc,
                              float* __restrict__ out) {
  out[0] = 1.0f - (float)(acc[0] / TOTAL_PIX);
}

extern "C" void kernel_launch(void* const* d_in, const int* in_sizes, int n_in,
                              void* d_out, int out_size, void* d_ws,
                              size_t ws_size, hipStream_t stream) {
  (void)in_sizes; (void)n_in; (void)out_size; (void)ws_size;
  const float* pred = (const float*)d_in[0];
  const float* targ = (const float*)d_in[1];
  float* out = (float*)d_out;
  double* acc = (double*)d_ws;

  hipMemsetAsync(d_ws, 0, sizeof(double), stream);

  dim3 grid(IMG_W / TILE_X, IMG_H / TILE_Y, NPLANES);  // (8, 32, 96)
  dim3 block(256);
  ssim_tile_kernel<<<grid, block, 0, stream>>>(pred, targ, acc);
  ssim_finalize<<<1, 1, 0, stream>>>(acc, out);
}